// HolographicInteractionLayer_5669356835978
// MI455X (gfx1250) — compile-verified
//
#include <hip/hip_runtime.h>

// Circular convolution (holographic interaction) via fp32 WMMA on gfx1250.
// out[b, r, n] = sum_k fe[b, p_r, k] * fe[b, q_r, (n-k) mod 64]
// Grouped by larger field q:  OUT_q (q x 64) = FE[0:q, :] @ Circulant(fe[q]).

typedef __attribute__((ext_vector_type(2))) float v2f;
typedef __attribute__((ext_vector_type(8))) float v8f;

#define NUM_FIELDS 39
#define NUM_PAIRS  741
#define EMB        64

__global__ __launch_bounds__(128)
void holo_interact_wmma_f32(const float* __restrict__ fe,
                            float* __restrict__ out,
                            int batch) {
  // A rows (up to 38 real rows; padded M-tile region up to row 47 stays in-bounds)
  __shared__ float ldsA[48 * EMB];   // 12 KB
  __shared__ float ldsB[2 * EMB];    // doubled row q for circulant indexing

  const int b   = blockIdx.x;
  const int q   = blockIdx.y + 1;      // q in 1..38 (uniform per block)
  const int tid = threadIdx.x;

  const float* feb = fe + (size_t)b * (NUM_FIELDS * EMB);

  // Stage A-rows 0..q-1 (contiguous prefix of this batch's block) into LDS (b128 loads).
  const float4* src4 = (const float4*)feb;
  const int n4 = q * (EMB / 4);
  for (int i = tid; i < n4; i += 128) ((float4*)ldsA)[i] = src4[i];
  // Doubled row q: ldsB[64 + n - k] == fe[q][(n-k) mod 64] for n,k in [0,64)
  ldsB[tid] = feb[q * EMB + (tid & 63)];
  __syncthreads();

  const int wave = tid >> 5;        // 4 waves: one 16-wide N strip each
  const int lane = tid & 31;
  const int half = lane >> 4;       // K sub-pair selector per WMMA A/B layout
  const int ln   = lane & 15;
  const int col  = (wave << 4) + ln;           // output column n

  // Circulant B fragment base: bb.x = bbase[-k0], bb.y = bbase[-k0-1]
  const float* bbase = &ldsB[64 + col - 2 * half];

  // Store-strip start: lane-half 0 begins at row p=0, half 1 at p=8.
  // Closed forms of the pair-index polynomial: pidx(0)=q-1, pidx(8)=267+q.
  const int pidx0 = half ? (267 + q) : (q - 1);
  float* addr = out + ((size_t)b * NUM_PAIRS + pidx0) * EMB + col;
  int base = half << 3;                        // this lane's strip starting row

  const int mTiles = (q + 15) >> 4;            // 1..3
  for (int mt = 0; mt < mTiles; ++mt) {
    v8f c = {};
    const float* arow = &ldsA[((mt << 4) + ln) * EMB];

#pragma unroll
    for (int k0 = 0; k0 < EMB; k0 += 4) {
      const int kb = k0 + 2 * half;            // lanes 0-15: k0,k0+1 ; 16-31: k0+2,k0+3
      v2f a;                                   // A 16x4 fragment (2 VGPRs)
      a.x = arow[kb];
      a.y = arow[kb + 1];
      v2f bb;                                  // B 4x16 circulant fragment
      bb.x = bbase[-k0];                       // C[kb  ][col]
      bb.y = bbase[-k0 - 1];                   // C[kb+1][col]
      c = __builtin_amdgcn_wmma_f32_16x16x4_f32(
              /*neg_a=*/false, a, /*neg_b=*/false, bb,
              /*c_mod=*/(short)0, c, /*reuse_a=*/false, /*reuse_b=*/false);
    }

    // ---- Store D tile: incremental pair-index addressing ----
    // VGPR j holds row base+j at column col; delta-pidx per row = 37-row.
    int d = (37 - base) * EMB;
    if (((mt + 1) << 4) <= q) {
      // Full tile (condition is SCALAR: mt,q uniform) -> no EXEC manipulation.
#pragma unroll
      for (int j = 0; j < 8; ++j) { *addr = c[j]; addr += d; d -= EMB; }
    } else {
      // Ragged tile: only ever the last iteration of a block.
#pragma unroll
      for (int j = 0; j < 8; ++j) {
        if (base + j < q) *addr = c[j];
        addr += d; d -= EMB;
      }
    }
    addr += (204 - 8 * base) * EMB;            // pidx(base+8) -> pidx(base+16)
    base += 16;
  }
}

extern "C" void kernel_launch(void* const* d_in, const int* in_sizes, int n_in,
                              void* d_out, int out_size, void* d_ws, size_t ws_size,
                              hipStream_t stream) {
  const float* fe = (const float*)d_in[0];
  float* out = (float*)d_out;
  const int batch = in_sizes[0] / (NUM_FIELDS * EMB);   // 4096

  dim3 grid(batch, NUM_FIELDS - 1);   // one workgroup per (batch, q)
  holo_interact_wmma_f32<<<grid, 128, 0, stream>>>(fe, out, batch);
}